// GCN_DeepSet_AntiSym_Invariant_30769145708787
// MI455X (gfx1250) — compile-verified
//
#include <hip/hip_runtime.h>
#include <hip/hip_bf16.h>
#include <math.h>

typedef __attribute__((ext_vector_type(2))) float v2f;
typedef __attribute__((ext_vector_type(8))) float v8f;

#define Bn  8
#define Nn  256
#define BN  2048
#define NID 64
#define Vd  128
#define GH  128
#define PHI 256
#define RHO 128

// ---------------------------------------------------------------------------
// Generic batched GEMM via V_WMMA_F32_16X16X4_F32.
// C[b] = act( rowscale(A[b]) @ B[b] + bias + addC[b] )
// One wave per 32x32 macro-tile (2x2 of 16x16 WMMA tiles): per K=4 step we
// load 2 A fragments (8B-aligned b64) + 2 B fragments and issue 4 WMMAs,
// halving vmem-per-wmma vs a 16x16 tile. All M,N multiples of 32 and grids
// exact multiples of 4 waves -> EXEC stays all-ones (WMMA requirement).
// ---------------------------------------------------------------------------
__global__ void __launch_bounds__(128)
gemm_wmma_f32(const float* __restrict__ A, long long sA, int lda,
              const float* __restrict__ Bm, long long sB, int ldb,
              const float* __restrict__ bias,
              const float* __restrict__ addC,
              const float* __restrict__ rowDiv,
              float* __restrict__ C, long long sC, int ldc,
              int M, int N, int K, int relu)
{
    const int lane   = threadIdx.x & 31;
    const int wave   = threadIdx.x >> 5;
    const int tilesN = N >> 5;                       // 32-wide macro tiles
    const int tile   = blockIdx.x * 4 + wave;
    if (tile >= (M >> 5) * tilesN) return;           // wave-uniform

    const int b  = blockIdx.z;
    const int m0 = (tile / tilesN) << 5;
    const int n0 = (tile % tilesN) << 5;

    const int r0   = m0 + (lane & 15);               // A rows held by this lane
    const int koff = (lane >> 4) << 1;               // K sub-offset per lane half
    const int c0   = n0 + (lane & 15);               // B/C cols held by this lane

    const float* A0 = A  + (size_t)b * (size_t)sA + (size_t)r0 * lda + koff;
    const float* A1 = A0 + (size_t)16 * lda;
    const float* B0 = Bm + (size_t)b * (size_t)sB + (size_t)koff * ldb + c0;
    const float* B1 = B0 + 16;

    float inv0 = 1.0f, inv1 = 1.0f;
    if (rowDiv) {                                    // mean aggregation: scale A rows
        inv0 = 1.0f / fmaxf(rowDiv[b * M + r0], 1.0f);
        inv1 = 1.0f / fmaxf(rowDiv[b * M + r0 + 16], 1.0f);
    }

    v8f acc00 = {0.f,0.f,0.f,0.f,0.f,0.f,0.f,0.f};
    v8f acc01 = acc00, acc10 = acc00, acc11 = acc00;

    for (int k = 0; k < K; k += 4) {
        v2f a0 = *(const v2f*)(A0 + k);              // 8B aligned -> global_load_b64
        v2f a1 = *(const v2f*)(A1 + k);
        a0 *= inv0;
        a1 *= inv1;
        const float* bp0 = B0 + (size_t)k * ldb;
        const float* bp1 = B1 + (size_t)k * ldb;
        v2f b0, b1;
        b0.x = bp0[0];  b0.y = bp0[ldb];
        b1.x = bp1[0];  b1.y = bp1[ldb];
        acc00 = __builtin_amdgcn_wmma_f32_16x16x4_f32(false, a0, false, b0, (short)0, acc00, false, false);
        acc01 = __builtin_amdgcn_wmma_f32_16x16x4_f32(false, a0, false, b1, (short)0, acc01, false, false);
        acc10 = __builtin_amdgcn_wmma_f32_16x16x4_f32(false, a1, false, b0, (short)0, acc10, false, false);
        acc11 = __builtin_amdgcn_wmma_f32_16x16x4_f32(false, a1, false, b1, (short)0, acc11, false, false);
    }

    const int crow = m0 + ((lane >> 4) << 3);        // base C row for VGPR v
    const float bv0 = bias ? bias[c0]      : 0.0f;
    const float bv1 = bias ? bias[c0 + 16] : 0.0f;

#pragma unroll
    for (int v = 0; v < 8; ++v) {
        size_t i00 = (size_t)b * (size_t)sC + (size_t)(crow + v)      * ldc + c0;
        size_t i01 = i00 + 16;
        size_t i10 = (size_t)b * (size_t)sC + (size_t)(crow + v + 16) * ldc + c0;
        size_t i11 = i10 + 16;
        float v00 = acc00[v] + bv0;
        float v01 = acc01[v] + bv1;
        float v10 = acc10[v] + bv0;
        float v11 = acc11[v] + bv1;
        if (addC) { v00 += addC[i00]; v01 += addC[i01]; v10 += addC[i10]; v11 += addC[i11]; }
        if (relu) {
            v00 = fmaxf(v00, 0.f); v01 = fmaxf(v01, 0.f);
            v10 = fmaxf(v10, 0.f); v11 = fmaxf(v11, 0.f);
        }
        C[i00] = v00; C[i01] = v01; C[i10] = v10; C[i11] = v11;
    }
}

// Build Mpos^T, Mneg^T, |A| and per-target relation counts from dense A.
__global__ void prep_masks_kernel(const float* __restrict__ A,
                                  float* __restrict__ mposT,
                                  float* __restrict__ mnegT,
                                  float* __restrict__ aabs,
                                  float* __restrict__ cnt0,
                                  float* __restrict__ cnt1)
{
    int t = blockIdx.x * blockDim.x + threadIdx.x;   // t = b*N + j (target)
    if (t >= Bn * Nn) return;
    int b = t >> 8, j = t & 255;
    const float* Ab = A + ((size_t)b << 16);
    float c0 = 0.f, c1 = 0.f;
    for (int i = 0; i < Nn; ++i) {
        float w = Ab[(i << 8) + j];
        float p = (w > 0.f) ? 1.f : 0.f;
        float q = (w < 0.f) ? 1.f : 0.f;
        mposT[((size_t)b << 16) + (j << 8) + i] = p;
        mnegT[((size_t)b << 16) + (j << 8) + i] = q;
        aabs [((size_t)b << 16) + (i << 8) + j] = fabsf(w);
        c0 += p; c1 += q;
    }
    cnt0[t] = c0;
    cnt1[t] = c1;
}

__global__ void zero_kernel(float* a, int n)
{
    int i = blockIdx.x * blockDim.x + threadIdx.x;
    if (i < n) a[i] = 0.f;
}

__global__ void add_inplace_kernel(float* __restrict__ a, const float* __restrict__ b, int n)
{
    int i = blockIdx.x * blockDim.x + threadIdx.x;
    if (i < n) a[i] += b[i];
}

// relu(LayerNorm(row)) over GH=128 features; one block (128 thr) per row.
__global__ void ln_relu_kernel(const float* __restrict__ in,
                               const float* __restrict__ g,
                               const float* __restrict__ bt,
                               float* __restrict__ out)
{
    __shared__ float red[128];
    const int row = blockIdx.x, t = threadIdx.x;
    float x = in[(size_t)row * GH + t];
    red[t] = x; __syncthreads();
    for (int s = 64; s > 0; s >>= 1) { if (t < s) red[t] += red[t + s]; __syncthreads(); }
    float mean = red[0] / (float)GH;
    __syncthreads();
    float d = x - mean;
    red[t] = d * d; __syncthreads();
    for (int s = 64; s > 0; s >>= 1) { if (t < s) red[t] += red[t + s]; __syncthreads(); }
    float var = red[0] / (float)GH;
    float y = d * rsqrtf(var + 1e-5f) * g[t] + bt[t];
    out[(size_t)row * GH + t] = fmaxf(y, 0.f);
}

// home/away masked sums of phi_out over nodes: one block per batch, thread = feature.
__global__ void masked_sum_kernel(const float* __restrict__ phi,
                                  const float* __restrict__ hm,
                                  float* __restrict__ hs,
                                  float* __restrict__ as_)
{
    int b = blockIdx.x, f = threadIdx.x;   // PHI = 256 threads
    float sh = 0.f, sa = 0.f;
    for (int n = 0; n < Nn; ++n) {
        float v = phi[((size_t)(b * Nn + n)) * PHI + f];
        float m = hm[b * Nn + n];
        sh += v * m;
        sa += v * (1.f - m);
    }
    hs[b * PHI + f] = sh;
    as_[b * PHI + f] = sa;
}

// rho MLP on home/away sums + antisymmetric tanh head. One block per batch.
__global__ void rho_final_kernel(const float* __restrict__ hs,
                                 const float* __restrict__ as_,
                                 const float* __restrict__ w1,
                                 const float* __restrict__ b1,
                                 const float* __restrict__ w2,
                                 const float* __restrict__ b2,
                                 float* __restrict__ out)
{
    __shared__ float sh[RHO], sa[RHO];
    int b = blockIdx.x, t = threadIdx.x;   // RHO = 128 threads
    float dh = 0.f, da = 0.f;
    for (int k = 0; k < PHI; ++k) {
        float w = w1[k * RHO + t];
        dh += hs[b * PHI + k] * w;
        da += as_[b * PHI + k] * w;
    }
    dh = fmaxf(dh + b1[t], 0.f);
    da = fmaxf(da + b1[t], 0.f);
    sh[t] = dh * w2[t];
    sa[t] = da * w2[t];
    __syncthreads();
    for (int s = 64; s > 0; s >>= 1) { if (t < s) { sh[t] += sh[t+s]; sa[t] += sa[t+s]; } __syncthreads(); }
    if (t == 0) {
        float o = (sh[0] + b2[0]) - (sa[0] + b2[0]);
        out[b] = 0.5f + 0.5f * tanhf(o);
    }
}

// ---------------------------------------------------------------------------

static void launch_gemm(const float* A, long long sA, int lda,
                        const float* Bm, long long sB, int ldb,
                        const float* bias, const float* addC, const float* rowDiv,
                        float* C, long long sC, int ldc,
                        int M, int N, int K, int relu, int batches, hipStream_t stream)
{
    int tiles = (M >> 5) * (N >> 5);                 // 32x32 macro tiles
    dim3 grid((tiles + 3) / 4, 1, batches);
    gemm_wmma_f32<<<grid, 128, 0, stream>>>(A, sA, lda, Bm, sB, ldb,
                                            bias, addC, rowDiv, C, sC, ldc,
                                            M, N, K, relu);
}

extern "C" void kernel_launch(void* const* d_in, const int* in_sizes, int n_in,
                              void* d_out, int out_size, void* d_ws, size_t ws_size,
                              hipStream_t stream)
{
    (void)in_sizes; (void)n_in; (void)out_size; (void)ws_size;

    const float* A        = (const float*)d_in[0];
    const float* X        = (const float*)d_in[1];
    const float* homeMask = (const float*)d_in[2];
    const float* emb1_w   = (const float*)d_in[3];
    const float* emb1_b   = (const float*)d_in[4];
    const float* emb2_w   = (const float*)d_in[5];
    const float* emb2_b   = (const float*)d_in[6];
    // per-block weights: [rgcn_w, rgcn_root, rgcn_bias, lina_w, lina_b, linb_w, linb_b]
    const float* blk[2][7] = {
        { (const float*)d_in[7],  (const float*)d_in[8],  (const float*)d_in[9],
          (const float*)d_in[10], (const float*)d_in[11], (const float*)d_in[12], (const float*)d_in[13] },
        { (const float*)d_in[14], (const float*)d_in[15], (const float*)d_in[16],
          (const float*)d_in[17], (const float*)d_in[18], (const float*)d_in[19], (const float*)d_in[20] },
    };
    const float* norm_g = (const float*)d_in[21];
    const float* norm_b = (const float*)d_in[22];
    const float* phi_w1 = (const float*)d_in[23];
    const float* phi_b1 = (const float*)d_in[24];
    const float* phi_w2 = (const float*)d_in[25];
    const float* phi_b2 = (const float*)d_in[26];
    const float* rho_w1 = (const float*)d_in[27];
    const float* rho_b1 = (const float*)d_in[28];
    const float* rho_w2 = (const float*)d_in[29];
    const float* rho_b2 = (const float*)d_in[30];
    float* out = (float*)d_out;

    // workspace layout (floats)
    float* ws    = (float*)d_ws;
    float* H     = ws;                    // 2048*128
    float* T1    = H     + BN * GH;       // 2048*256
    float* T2    = T1    + BN * PHI;      // 2048*256
    float* agg   = T2    + BN * PHI;      // 2048*128
    float* H2    = agg   + BN * GH;       // 2048*128
    float* mposT = H2    + BN * GH;       // 8*256*256
    float* mnegT = mposT + Bn * Nn * Nn;  // 8*256*256
    float* aabs  = mnegT + Bn * Nn * Nn;  // 8*256*256
    float* cnt0  = aabs  + Bn * Nn * Nn;  // 2048
    float* cnt1  = cnt0  + BN;            // 2048
    float* hsum  = cnt1  + BN;            // 8*256
    float* asum  = hsum  + Bn * PHI;      // 8*256

    const long long sBat = (long long)Nn * GH;       // 256*128 per-batch stride
    const long long sM   = (long long)Nn * Nn;       // 256*256 mask stride

    // ---- embeddings: H = (relu(X@W1+b1))@W2 + b2 ----
    launch_gemm(X, 0, NID, emb1_w, 0, Vd, emb1_b, nullptr, nullptr,
                T1, 0, Vd, BN, Vd, NID, 1, 1, stream);
    launch_gemm(T1, 0, Vd, emb2_w, 0, Vd, emb2_b, nullptr, nullptr,
                H, 0, Vd, BN, Vd, Vd, 0, 1, stream);

    // ---- edge-derived dense operators ----
    prep_masks_kernel<<<(Bn * Nn + 255) / 256, 256, 0, stream>>>(A, mposT, mnegT, aabs, cnt0, cnt1);
    zero_kernel<<<(BN * GH + 255) / 256, 256, 0, stream>>>(agg, BN * GH);

    for (int k = 0; k < 2; ++k) {
        const float* Wrel0 = blk[k][0];
        const float* Wrel1 = blk[k][0] + Vd * GH;
        const float* Wroot = blk[k][1];
        const float* rbias = blk[k][2];
        const float* law   = blk[k][3];
        const float* lab   = blk[k][4];
        const float* lbw   = blk[k][5];
        const float* lbb   = blk[k][6];

        // H = H + agg
        add_inplace_kernel<<<(BN * GH + 255) / 256, 256, 0, stream>>>(H, agg, BN * GH);

        // S_r = M_r^T @ H   (batched 256x256 @ 256x128)
        launch_gemm(mposT, sM, Nn, H, sBat, GH, nullptr, nullptr, nullptr,
                    T1, sBat, GH, Nn, GH, Nn, 0, Bn, stream);
        launch_gemm(mnegT, sM, Nn, H, sBat, GH, nullptr, nullptr, nullptr,
                    T2, sBat, GH, Nn, GH, Nn, 0, Bn, stream);

        // H2 = H @ Wroot + bias
        launch_gemm(H, 0, GH, Wroot, 0, GH, rbias, nullptr, nullptr,
                    H2, 0, GH, BN, GH, GH, 0, 1, stream);
        // H2 += (S0 / max(c0,1)) @ Wrel0   ;   H2 += (S1 / max(c1,1)) @ Wrel1
        launch_gemm(T1, sBat, GH, Wrel0, 0, GH, nullptr, H2, cnt0,
                    H2, sBat, GH, Nn, GH, GH, 0, Bn, stream);
        launch_gemm(T2, sBat, GH, Wrel1, 0, GH, nullptr, H2, cnt1,
                    H2, sBat, GH, Nn, GH, GH, 0, Bn, stream);

        // agg_raw = |A| @ H2   (batched)
        launch_gemm(aabs, sM, Nn, H2, sBat, GH, nullptr, nullptr, nullptr,
                    T1, sBat, GH, Nn, GH, Nn, 0, Bn, stream);

        // agg = relu(LN(agg_raw)) ; relu(@law+lab) ; @lbw+lbb
        ln_relu_kernel<<<BN, 128, 0, stream>>>(T1, norm_g, norm_b, T2);
        launch_gemm(T2, 0, GH, law, 0, GH, lab, nullptr, nullptr,
                    T1, 0, GH, BN, GH, GH, 1, 1, stream);
        launch_gemm(T1, 0, GH, lbw, 0, GH, lbb, nullptr, nullptr,
                    agg, 0, GH, BN, GH, GH, 0, 1, stream);
    }

    // H = H + agg (final residual)
    add_inplace_kernel<<<(BN * GH + 255) / 256, 256, 0, stream>>>(H, agg, BN * GH);

    // ---- DeepSet phi ----
    launch_gemm(H, 0, GH, phi_w1, 0, PHI, phi_b1, nullptr, nullptr,
                T1, 0, PHI, BN, PHI, GH, 1, 1, stream);
    launch_gemm(T1, 0, PHI, phi_w2, 0, PHI, phi_b2, nullptr, nullptr,
                T2, 0, PHI, BN, PHI, PHI, 1, 1, stream);

    masked_sum_kernel<<<Bn, PHI, 0, stream>>>(T2, homeMask, hsum, asum);
    rho_final_kernel<<<Bn, RHO, 0, stream>>>(hsum, asum, rho_w1, rho_b1, rho_w2, rho_b2, out);
}